// SumLayer_37787122270369
// MI455X (gfx1250) — compile-verified
//
#include <hip/hip_runtime.h>
#include <math.h>

#define BATCH   256
#define PART    128
#define ONODES  64
#define INODES  64
#define BTILE   64
#define LDS_STRIDE 68   // 64 + 4 pad: float4-aligned, bank-conflict-free for WMMA reads

typedef __attribute__((ext_vector_type(2))) float v2f;
typedef __attribute__((ext_vector_type(8))) float v8f;

// Low 32 bits of a generic pointer to LDS == DS byte address (ISA 10.2 aperture map)
__device__ __forceinline__ unsigned lds_addr32(const void* p) {
    return (unsigned)(uintptr_t)p;
}

__global__ __launch_bounds__(256)
void sumlayer_wmma_f32(const float* __restrict__ x,
                       const float* __restrict__ weight,
                       float* __restrict__ out) {
    __shared__ float sS[ONODES * LDS_STRIDE];  // softmax(weight[p]) : S[o][i]
    __shared__ float sE[BTILE  * LDS_STRIDE];  // exp(x - rowmax)    : E[b][i]
    __shared__ float sM[BTILE];                // per-batch-row max

    const int t  = threadIdx.x;
    const int p  = blockIdx.y;
    const int b0 = blockIdx.x * BTILE;

    const float* wp = weight + (size_t)p * ONODES * INODES;
    const float* xp = x + ((size_t)b0 * PART + p) * INODES;

    // ---- async global -> LDS (no VGPR staging), 16B per lane per issue ----
    #pragma unroll
    for (int it = 0; it < 4; ++it) {
        const int idx = t + 256 * it;            // float4 index 0..1023
        const int row = idx >> 4;                // 16 float4 per 64-float row
        const int c4  = (idx & 15) << 2;
        const unsigned sOffW = lds_addr32(&sS[row * LDS_STRIDE + c4]);
        const unsigned sOffX = lds_addr32(&sE[row * LDS_STRIDE + c4]);
        const unsigned gOffW = (unsigned)((row * INODES + c4) * sizeof(float));
        const unsigned gOffX = (unsigned)((row * (PART * INODES) + c4) * sizeof(float));
        asm volatile("global_load_async_to_lds_b128 %0, %1, %2"
                     :: "v"(sOffW), "v"(gOffW), "s"(wp) : "memory");
        asm volatile("global_load_async_to_lds_b128 %0, %1, %2"
                     :: "v"(sOffX), "v"(gOffX), "s"(xp) : "memory");
    }
    asm volatile("s_wait_asynccnt 0" ::: "memory");
    __syncthreads();

    // ---- row transforms: 4 lanes/row, 16 elems/lane, shfl_xor group reduce ----
    {
        const int row = t >> 2;        // 0..63
        const int q   = (t & 3) * 16;  // column base

        // softmax of weight row (in place in sS)
        {
            float* r = &sS[row * LDS_STRIDE + q];
            float v[16];
            #pragma unroll
            for (int i = 0; i < 16; ++i) v[i] = r[i];
            float mx = v[0];
            #pragma unroll
            for (int i = 1; i < 16; ++i) mx = fmaxf(mx, v[i]);
            mx = fmaxf(mx, __shfl_xor(mx, 1, 32));
            mx = fmaxf(mx, __shfl_xor(mx, 2, 32));
            float sum = 0.f;
            #pragma unroll
            for (int i = 0; i < 16; ++i) { v[i] = __expf(v[i] - mx); sum += v[i]; }
            sum += __shfl_xor(sum, 1, 32);
            sum += __shfl_xor(sum, 2, 32);
            const float inv = 1.f / sum;
            #pragma unroll
            for (int i = 0; i < 16; ++i) r[i] = v[i] * inv;
        }
        // exp-with-max of x row (in place in sE), record row max
        {
            float* e = &sE[row * LDS_STRIDE + q];
            float v[16];
            #pragma unroll
            for (int i = 0; i < 16; ++i) v[i] = e[i];
            float mx = v[0];
            #pragma unroll
            for (int i = 1; i < 16; ++i) mx = fmaxf(mx, v[i]);
            mx = fmaxf(mx, __shfl_xor(mx, 1, 32));
            mx = fmaxf(mx, __shfl_xor(mx, 2, 32));
            if ((t & 3) == 0) sM[row] = mx;
            #pragma unroll
            for (int i = 0; i < 16; ++i) e[i] = __expf(v[i] - mx);
        }
    }
    __syncthreads();

    // ---- WMMA: out_tile(64x64) = E(64x64) @ S^T(64x64), fp32 ----
    const int lane = t & 31;
    const int wave = t >> 5;       // 8 waves
    const int lo   = lane & 15;
    const int hi   = lane >> 4;

    #pragma unroll
    for (int st = 0; st < 2; ++st) {
        const int sub = wave * 2 + st;      // 16 sub-tiles of 16x16
        const int m0  = (sub >> 2) * 16;    // batch-tile row
        const int n0  = (sub & 3)  * 16;    // out-node col
        v8f c = {0.f, 0.f, 0.f, 0.f, 0.f, 0.f, 0.f, 0.f};

        const float* aRow = &sE[(m0 + lo) * LDS_STRIDE];  // A[m][k], m = lane&15
        const float* bRow = &sS[(n0 + lo) * LDS_STRIDE];  // B[k][n] = S[n][k]

        #pragma unroll
        for (int k0 = 0; k0 < INODES; k0 += 4) {
            const int kk = k0 + 2 * hi;     // ISA 16x4 f32 layout: vgpr=k&1, lanehalf=k>>1
            v2f a, b;
            a.x = aRow[kk]; a.y = aRow[kk + 1];
            b.x = bRow[kk]; b.y = bRow[kk + 1];
            // (neg_a, A, neg_b, B, c_mod, C, reuse_a, reuse_b)
            c = __builtin_amdgcn_wmma_f32_16x16x4_f32(
                    false, a, false, b, (short)0, c, false, false);
        }

        // C/D layout: VGPR r, lanes 0-15 -> M=r, lanes 16-31 -> M=r+8; N = lane&15
        #pragma unroll
        for (int r = 0; r < 8; ++r) {
            const int m  = m0 + r + 8 * hi;
            const int n  = n0 + lo;
            const int bg = b0 + m;
            const float val = __logf(c[r]) + sM[m];
            out[((size_t)bg * PART + p) * ONODES + n] = val;
        }
    }
}

extern "C" void kernel_launch(void* const* d_in, const int* in_sizes, int n_in,
                              void* d_out, int out_size, void* d_ws, size_t ws_size,
                              hipStream_t stream) {
    const float* x = (const float*)d_in[0];
    const float* w = (const float*)d_in[1];
    float* out     = (float*)d_out;
    (void)in_sizes; (void)n_in; (void)out_size; (void)d_ws; (void)ws_size;

    dim3 grid(BATCH / BTILE, PART);   // 4 x 128 = 512 workgroups
    sumlayer_wmma_f32<<<grid, 256, 0, stream>>>(x, w, out);
}